// QuantumGenerator_34394098106454
// MI455X (gfx1250) — compile-verified
//
#include <hip/hip_runtime.h>
#include <math.h>

#define NQ      16
#define DIM     65536      // 2^16 state amplitudes (real)
#define QT      1024       // threads per state-sim block (32 wave32)
#define NLAYERS 6
#define BATCH   256
#define OUTD    256

typedef __attribute__((ext_vector_type(2))) float v2f;
typedef __attribute__((ext_vector_type(8))) float v8f;

// ---------------------------------------------------------------------------
// Kernel 1: full state-vector simulation, one sample per workgroup.
// State (256 KB) lives entirely in LDS (MI455X WGP has 320 KB).
// ---------------------------------------------------------------------------

__device__ __forceinline__ void apply_ry(float* st, int w, float theta, int tid) {
    float s, c;
    __sincosf(0.5f * theta, &s, &c);
    const int bit  = 1 << w;
    const int mlow = bit - 1;
#pragma unroll 4
    for (int p = tid; p < (DIM >> 1); p += QT) {
        const int i0 = ((p >> w) << (w + 1)) | (p & mlow);
        const int i1 = i0 | bit;
        const float a0 = st[i0];
        const float a1 = st[i1];
        st[i0] = c * a0 - s * a1;
        st[i1] = s * a0 + c * a1;
    }
    __syncthreads();
}

__device__ __forceinline__ void apply_cnot(float* st, int ctrl, int tgt, int tid) {
    const int lo = ctrl < tgt ? ctrl : tgt;
    const int hi = ctrl < tgt ? tgt : ctrl;
    const int cb = 1 << ctrl;
    const int tb = 1 << tgt;
#pragma unroll 4
    for (int q = tid; q < (DIM >> 2); q += QT) {
        int i = ((q >> lo) << (lo + 1)) | (q & ((1 << lo) - 1));
        i     = ((i >> hi) << (hi + 1)) | (i & ((1 << hi) - 1));
        const int i0 = i | cb;          // ctrl=1, tgt=0
        const int i1 = i0 | tb;         // ctrl=1, tgt=1
        const float a = st[i0];
        const float b = st[i1];
        st[i0] = b;
        st[i1] = a;
    }
    __syncthreads();
}

__global__ __launch_bounds__(QT)
void qg_state_kernel(const float* __restrict__ z,
                     const float* __restrict__ qw,
                     float* __restrict__ qout) {
    extern __shared__ float smem[];
    float* st    = smem;          // DIM floats
    float* accum = smem + DIM;    // NQ floats

    const int tid = threadIdx.x;
    const int b   = blockIdx.x;

    // init |0...0>
    for (int i = tid; i < DIM; i += QT) st[i] = 0.0f;
    if (tid == 0)  st[0] = 1.0f;
    if (tid < NQ)  accum[tid] = 0.0f;
    __syncthreads();

    // Fused AngleEmbedding RY(z) + layer-0 RY(theta0): RY(a)RY(b)=RY(a+b)
    for (int w = 0; w < NQ; ++w)
        apply_ry(st, w, z[b * NQ + w] + qw[w], tid);
    for (int w = 0; w < NQ; ++w)
        apply_cnot(st, w, (w + 1) & (NQ - 1), tid);

    for (int l = 1; l < NLAYERS; ++l) {
        for (int w = 0; w < NQ; ++w)
            apply_ry(st, w, qw[l * NQ + w], tid);
        for (int w = 0; w < NQ; ++w)
            apply_cnot(st, w, (w + 1) & (NQ - 1), tid);
    }

    // <Z_w> = sum_i |a_i|^2 * (bit_w(i) ? -1 : +1)
    float acc[NQ];
#pragma unroll
    for (int w = 0; w < NQ; ++w) acc[w] = 0.0f;
    for (int i = tid; i < DIM; i += QT) {
        const float a = st[i];
        const float p = a * a;                       // >= 0
#pragma unroll
        for (int w = 0; w < NQ; ++w) {
            const int sgn = (i << (31 - w)) & 0x80000000;  // sign flip if bit set
            acc[w] += __int_as_float(__float_as_int(p) ^ sgn);
        }
    }

    // wave32 reduction, then LDS float atomics (ds_add_f32)
    const int lane = tid & 31;
#pragma unroll
    for (int w = 0; w < NQ; ++w) {
        float v = acc[w];
#pragma unroll
        for (int off = 16; off; off >>= 1) v += __shfl_xor(v, off, 32);
        if (lane == 0) atomicAdd(&accum[w], v);
    }
    __syncthreads();
    if (tid < NQ) qout[b * NQ + tid] = accum[tid];
}

// ---------------------------------------------------------------------------
// Kernel 2: out = tanh(q @ W^T + b) via V_WMMA_F32_16X16X4_F32.
// One wave per 16x16 output tile; K=16 as 4 chained k=4 WMMAs.
// A (16x4 f32): lanes 0-15 -> K{0,1}, lanes 16-31 -> K{2,3} (VGPR0=even K).
// B (4x16 f32): VGPR v holds rows K=v (lanes 0-15) / K=v+2 (lanes 16-31).
// C/D (16x16 f32): VGPR v, lane L -> M = v + (L>=16)*8, N = L%16.
// ---------------------------------------------------------------------------

__global__ __launch_bounds__(256)
void qg_head_wmma_kernel(const float* __restrict__ q,
                         const float* __restrict__ W,
                         const float* __restrict__ bias,
                         float* __restrict__ out) {
    const int lane = threadIdx.x & 31;
    const int wave = threadIdx.x >> 5;
    const int t    = blockIdx.x * 8 + wave;   // tile id 0..255
    const int tm   = (t >> 4) << 4;           // batch-row base
    const int tn   = (t & 15) << 4;           // out-col base
    const int half = lane >> 4;               // 0: lanes 0-15, 1: lanes 16-31
    const int mr   = lane & 15;               // M (for A) == N (for B/C/D)

    v8f c = {};
#pragma unroll
    for (int kk = 0; kk < 4; ++kk) {
        const int kb = kk * 4 + half * 2;
        v2f a, bm;
        a.x  = q[(tm + mr) * NQ + kb];        // A[M=mr][K=kb]
        a.y  = q[(tm + mr) * NQ + kb + 1];    // A[M=mr][K=kb+1]
        bm.x = W[(tn + mr) * NQ + kb];        // B[K=kb][N=mr] = W[n][k]
        bm.y = W[(tn + mr) * NQ + kb + 1];
        c = __builtin_amdgcn_wmma_f32_16x16x4_f32(
                /*neg_a=*/false, a, /*neg_b=*/false, bm,
                /*c_mod=*/(short)0, c, /*reuse_a=*/false, /*reuse_b=*/false);
    }

    const float bv = bias[tn + mr];
#pragma unroll
    for (int v = 0; v < 8; ++v) {
        const int m = v + half * 8;
        out[(tm + m) * OUTD + tn + mr] = tanhf(c[v] + bv);
    }
}

// ---------------------------------------------------------------------------

extern "C" void kernel_launch(void* const* d_in, const int* in_sizes, int n_in,
                              void* d_out, int out_size, void* d_ws, size_t ws_size,
                              hipStream_t stream) {
    const float* z    = (const float*)d_in[0];  // (256, 16)
    const float* qw   = (const float*)d_in[1];  // (6, 16)
    const float* W    = (const float*)d_in[2];  // (256, 16)
    const float* bias = (const float*)d_in[3];  // (256,)
    float* out  = (float*)d_out;                // (256, 256)
    float* qout = (float*)d_ws;                 // (256, 16) scratch

    const size_t shmem = (size_t)(DIM + NQ) * sizeof(float);  // 256 KB + 64 B
    qg_state_kernel<<<BATCH, QT, shmem, stream>>>(z, qw, qout);

    // 256 tiles of 16x16, 8 waves (tiles) per 256-thread block -> 32 blocks
    qg_head_wmma_kernel<<<(BATCH / 16) * (OUTD / 16) / 8, 256, 0, stream>>>(
        qout, W, bias, out);
}